// GraphUNet_48258252538508
// MI455X (gfx1250) — compile-verified
//
#include <hip/hip_runtime.h>
#include <math.h>

// ---------------------------------------------------------------------------
// Graph U-Net forward for MI455X (gfx1250, wave32).
// Heavy math: f32 WMMA (V_WMMA_F32_16X16X4_F32) tiled GEMMs with
// double-buffered LDS fed by GLOBAL_LOAD_ASYNC_TO_LDS_B128 (ASYNCcnt).
// ---------------------------------------------------------------------------

typedef __attribute__((ext_vector_type(2))) float v2f;
typedef __attribute__((ext_vector_type(8))) float v8f;

enum { EPI_PLAIN = 0, EPI_ROWSCALE = 1, EPI_GCN = 2, EPI_GCN_RELU = 3, EPI_AUG = 4 };

// Async 16B copy global -> LDS (no VGPR round trip, tracked by ASYNCcnt).
// LDS operand: low 32 bits of the flat address == LDS byte offset (ISA 10.2).
__device__ __forceinline__ void async_cp16(const float* g, float* l) {
    unsigned loff = (unsigned)(size_t)l;
    unsigned long long ga = (unsigned long long)(size_t)g;
    asm volatile("global_load_async_to_lds_b128 %0, %1, off"
                 :: "v"(loff), "v"(ga)
                 : "memory");
}

// ---------------------------------------------------------------------------
// WMMA f32 GEMM: C = epilogue(A[MxK] @ B[KxN])
// Requirements: M%128==0, N%128==0, K%16==0 (true for every GEMM here).
// Block: 256 threads (8 waves). Workgroup tile 128x128, BK=16.
// Wave layout: wm=wid&3 (4 rows of 32), wn=wid>>2 (2 cols of 64).
// Each wave: 2x4 grid of 16x16 WMMA tiles -> acc[2][4] of v8f.
// LDS: A tile [128][20] (b64 frag reads, conflict-free), B tile row-major
// [16][136] (2xb32 frag reads; stride 136 -> the 16-lane groups hit disjoint
// bank ranges). Double buffered; staging via async-to-LDS b128, 4 per thread
// per stage, overlapped with compute of the previous stage.
// ---------------------------------------------------------------------------
template <int EPI>
__global__ __launch_bounds__(256) void wmma_gemm_f32(
    const float* __restrict__ A, const float* __restrict__ B,
    float* __restrict__ C, int M, int N, int K,
    const float* __restrict__ dinv,   // EPI_ROWSCALE / EPI_GCN*
    const float* __restrict__ Zaux,   // EPI_GCN*: z (MxN); EPI_AUG: A (MxN)
    const float* __restrict__ bias)   // EPI_GCN*
{
    constexpr int BM = 128, BN = 128, BK = 16;
    constexpr int LDA_ = 20;   // A tile row stride (floats): 80B, 16B aligned
    constexpr int LDB_ = 136;  // B tile row stride (floats): 544B, 16B aligned
    __shared__ __align__(16) float Alds[2][BM * LDA_];
    __shared__ __align__(16) float Blds[2][BK * LDB_];

    const int tid = threadIdx.x;
    const int bm0 = blockIdx.y * BM;
    const int bn0 = blockIdx.x * BN;

    const int lane = tid & 31;
    const int wid  = tid >> 5;
    const int wm = wid & 3;    // wave row  -> rows  [wm*32, wm*32+32)
    const int wn = wid >> 2;   // wave col  -> cols  [wn*64, wn*64+64)
    const int lr = lane & 15;  // lane within half
    const int lh = lane >> 4;  // half select (K sub-pair / M+8 for C)

    v8f acc[2][4];
    const v8f vzero = {0.f, 0.f, 0.f, 0.f, 0.f, 0.f, 0.f, 0.f};
#pragma unroll
    for (int i = 0; i < 2; i++)
#pragma unroll
        for (int j = 0; j < 4; j++) acc[i][j] = vzero;

    // Issue one stage of async tile loads (A: 2 chunks, B: 2 chunks / thread).
    auto stage = [&](int kt, int b) {
#pragma unroll
        for (int s = 0; s < 2; s++) {
            int f   = tid + s * 256;   // 0..511
            int row = f >> 2;          // 0..127
            int c4  = (f & 3) << 2;    // 0,4,8,12
            async_cp16(&A[(size_t)(bm0 + row) * K + kt + c4],
                       &Alds[b][row * LDA_ + c4]);
        }
#pragma unroll
        for (int s = 0; s < 2; s++) {
            int f   = tid + s * 256;
            int row = f >> 5;          // k row 0..15
            int c4  = (f & 31) << 2;   // n col 0..124
            async_cp16(&B[(size_t)(kt + row) * N + bn0 + c4],
                       &Blds[b][row * LDB_ + c4]);
        }
    };

    stage(0, 0);  // prologue: stage 0 in flight

    int buf = 0;
    for (int kt = 0; kt < K; kt += BK, buf ^= 1) {
        __syncthreads();  // all waves done reading the buffer we now overwrite
        const bool more = (kt + BK) < K;
        if (more) {
            stage(kt + BK, buf ^ 1);  // overlap next-stage loads with compute
            asm volatile("s_wait_asynccnt 0x4" ::: "memory");  // stage kt done
        } else {
            asm volatile("s_wait_asynccnt 0x0" ::: "memory");
        }
        __syncthreads();  // stage kt visible to every wave

        const float* Ab = &Alds[buf][0];
        const float* Bb = &Blds[buf][0];
#pragma unroll
        for (int ks = 0; ks < BK / 4; ks++) {
            const int kb = ks * 4 + lh * 2;  // this lane's K pair
            v2f a[2], b[4];
            a[0] = *(const v2f*)&Ab[(wm * 32 + lr) * LDA_ + kb];
            a[1] = *(const v2f*)&Ab[(wm * 32 + 16 + lr) * LDA_ + kb];
#pragma unroll
            for (int t = 0; t < 4; t++) {
                const int nn = wn * 64 + t * 16 + lr;
                v2f bt;
                bt.x = Bb[kb * LDB_ + nn];
                bt.y = Bb[(kb + 1) * LDB_ + nn];
                b[t] = bt;
            }
#pragma unroll
            for (int mt = 0; mt < 2; mt++)
#pragma unroll
                for (int nt = 0; nt < 4; nt++)
                    acc[mt][nt] = __builtin_amdgcn_wmma_f32_16x16x4_f32(
                        false, a[mt], false, b[nt], (short)0, acc[mt][nt],
                        false, false);
        }
    }

    // Epilogue. C/D layout: element v of v8f at lane l -> M = v + 8*(l>>4),
    // N = l&15 within the 16x16 tile.
#pragma unroll
    for (int mt = 0; mt < 2; mt++) {
#pragma unroll
        for (int nt = 0; nt < 4; nt++) {
#pragma unroll
            for (int v = 0; v < 8; v++) {
                int gi = bm0 + wm * 32 + mt * 16 + lh * 8 + v;
                int gj = bn0 + wn * 64 + nt * 16 + lr;
                size_t o = (size_t)gi * N + gj;
                float r = acc[mt][nt][v];
                if (EPI == EPI_PLAIN) {
                    C[o] = r;
                } else if (EPI == EPI_ROWSCALE) {
                    C[o] = r * dinv[gi];
                } else if (EPI == EPI_GCN || EPI == EPI_GCN_RELU) {
                    float y = dinv[gi] * (r + 2.0f * Zaux[o]) + bias[gj];
                    if (EPI == EPI_GCN_RELU) y = fmaxf(y, 0.0f);
                    C[o] = y;
                } else {  // EPI_AUG: (A+I)^2 with diag zeroed, diag(A)=0
                    C[o] = (gi == gj) ? 0.0f : (r + 2.0f * Zaux[o]);
                }
            }
        }
    }
}

// deg[j] = colsum(A)[j] + 2 (improved self loop); dinv = rsqrt(deg)
__global__ void colsum_dinv_kernel(const float* __restrict__ A, int n,
                                   float* __restrict__ dinv) {
    int j = blockIdx.x * blockDim.x + threadIdx.x;
    if (j >= n) return;
    float s = 2.0f;
    for (int i = 0; i < n; i++) s += A[(size_t)i * n + j];
    dinv[j] = (s > 0.0f) ? rsqrtf(s) : 0.0f;
}

// score[i] = tanh( (x[i] . p) / ||p|| )
__global__ void score_kernel(const float* __restrict__ x,
                             const float* __restrict__ p, int n, int H,
                             float* __restrict__ score) {
    __shared__ float s_inv;
    if (threadIdx.x == 0) {
        float s = 0.f;
        for (int c = 0; c < H; c++) s += p[c] * p[c];
        s_inv = rsqrtf(s);
    }
    __syncthreads();
    int i = blockIdx.x * blockDim.x + threadIdx.x;
    if (i >= n) return;
    float d = 0.f;
    const float* xr = x + (size_t)i * H;
    for (int c = 0; c < H; c++) d += xr[c] * p[c];
    score[i] = tanhf(d * s_inv);
}

// Single-workgroup bitonic top-k (descending, index-ascending tie break).
// n is a power of two <= 4096.
__global__ __launch_bounds__(1024) void topk_kernel(
    const float* __restrict__ score, int n, int k, int* __restrict__ perm,
    float* __restrict__ sval) {
    __shared__ float skey[4096];
    __shared__ int   sidx[4096];
    const int tid = threadIdx.x;
    for (int i = tid; i < n; i += 1024) { skey[i] = score[i]; sidx[i] = i; }
    __syncthreads();
    for (int size = 2; size <= n; size <<= 1) {
        for (int stride = size >> 1; stride > 0; stride >>= 1) {
            for (int i = tid; i < (n >> 1); i += 1024) {
                int pos = 2 * i - (i & (stride - 1));
                int j = pos + stride;
                bool descBlock = ((pos & size) == 0);
                float ka = skey[pos], kb = skey[j];
                int ia = sidx[pos], ib = sidx[j];
                bool ab = (ka > kb) || (ka == kb && ia < ib);  // a before b
                if (descBlock ? (!ab) : ab) {
                    skey[pos] = kb; skey[j] = ka;
                    sidx[pos] = ib; sidx[j] = ia;
                }
            }
            __syncthreads();
        }
    }
    for (int i = tid; i < k; i += 1024) { perm[i] = sidx[i]; sval[i] = skey[i]; }
}

// out[r,:] = x[perm[r],:] * sval[r]
__global__ void gather_rows_kernel(const float* __restrict__ x,
                                   const int* __restrict__ perm,
                                   const float* __restrict__ sval, int H, int k,
                                   float* __restrict__ out) {
    int t = blockIdx.x * blockDim.x + threadIdx.x;
    if (t >= k * H) return;
    int r = t / H, c = t % H;
    out[t] = x[(size_t)perm[r] * H + c] * sval[r];
}

// Ap[r,c] = Aaug[perm[r], perm[c]]
__global__ void gather_adj_kernel(const float* __restrict__ Aaug, int n,
                                  const int* __restrict__ perm, int k,
                                  float* __restrict__ Ap) {
    size_t t = (size_t)blockIdx.x * blockDim.x + threadIdx.x;
    if (t >= (size_t)k * k) return;
    int r = (int)(t / k), c = (int)(t % k);
    Ap[t] = Aaug[(size_t)perm[r] * n + perm[c]];
}

// cat[i, 0:H] = res[i, :]; cat[i, H:2H] = 0
__global__ void concat_init_kernel(const float* __restrict__ res, int n, int H,
                                   float* __restrict__ cat) {
    size_t t = (size_t)blockIdx.x * blockDim.x + threadIdx.x;
    if (t >= (size_t)n * 2 * H) return;
    int i = (int)(t / (2 * H)), c = (int)(t % (2 * H));
    cat[t] = (c < H) ? res[(size_t)i * H + c] : 0.0f;
}

// cat[perm[r], H + c] = x[r, c]
__global__ void scatter_rows_kernel(const float* __restrict__ x,
                                    const int* __restrict__ perm, int k, int H,
                                    float* __restrict__ cat) {
    int t = blockIdx.x * blockDim.x + threadIdx.x;
    if (t >= k * H) return;
    int r = t / H, c = t % H;
    cat[(size_t)perm[r] * (2 * H) + H + c] = x[(size_t)r * H + c];
}

// ---------------------------------------------------------------------------
// Host orchestration
// ---------------------------------------------------------------------------
static inline void launch_gcn(hipStream_t stream, const float* Aadj, int n,
                              const float* X, int Kin, int Nout, const float* W,
                              const float* bias, float* out, bool relu,
                              float* dinv, float* z) {
    colsum_dinv_kernel<<<(n + 255) / 256, 256, 0, stream>>>(Aadj, n, dinv);
    dim3 g1(Nout / 128, n / 128);
    wmma_gemm_f32<EPI_ROWSCALE><<<g1, 256, 0, stream>>>(X, W, z, n, Nout, Kin,
                                                        dinv, nullptr, nullptr);
    dim3 g2(Nout / 128, n / 128);
    if (relu)
        wmma_gemm_f32<EPI_GCN_RELU><<<g2, 256, 0, stream>>>(Aadj, z, out, n,
                                                            Nout, n, dinv, z,
                                                            bias);
    else
        wmma_gemm_f32<EPI_GCN><<<g2, 256, 0, stream>>>(Aadj, z, out, n, Nout, n,
                                                       dinv, z, bias);
}

static inline void launch_augment(hipStream_t stream, const float* Ain, int n,
                                  float* Aaug) {
    dim3 g(n / 128, n / 128);
    wmma_gemm_f32<EPI_AUG><<<g, 256, 0, stream>>>(Ain, Ain, Aaug, n, n, n,
                                                  nullptr, Ain, nullptr);
}

extern "C" void kernel_launch(void* const* d_in, const int* in_sizes, int n_in,
                              void* d_out, int out_size, void* d_ws,
                              size_t ws_size, hipStream_t stream) {
    (void)in_sizes; (void)n_in; (void)out_size; (void)ws_size;
    const int N0 = 4096, H = 256;

    const float* x_in = (const float*)d_in[0];   // 4096 x 128
    const float* A_in = (const float*)d_in[1];   // 4096 x 4096
    const float* Wd0 = (const float*)d_in[2];  const float* bd0 = (const float*)d_in[3];
    const float* Wd1 = (const float*)d_in[4];  const float* bd1 = (const float*)d_in[5];
    const float* Wd2 = (const float*)d_in[6];  const float* bd2 = (const float*)d_in[7];
    const float* Wd3 = (const float*)d_in[8];  const float* bd3 = (const float*)d_in[9];
    const float* p1  = (const float*)d_in[10];
    const float* p2  = (const float*)d_in[11];
    const float* p3  = (const float*)d_in[12];
    const float* Wu0 = (const float*)d_in[13]; const float* bu0 = (const float*)d_in[14];
    const float* Wu1 = (const float*)d_in[15]; const float* bu1 = (const float*)d_in[16];
    const float* Wu2 = (const float*)d_in[17]; const float* bu2 = (const float*)d_in[18];

    float* out_x = (float*)d_out;                         // 4096 x 128 result
    float* Aaug  = (float*)d_out + (size_t)N0 * 128;      // n x n scratch, then final A

    // Workspace carve-out (floats).
    float* ws = (float*)d_ws;
    size_t off = 0;
    auto alloc = [&](size_t nfl) { float* p = ws + off; off += nfl; return p; };
    float* dinv = alloc(N0);
    float* score = alloc(N0);
    float* sval = alloc(N0);
    float* z    = alloc((size_t)N0 * H);
    float* xs0  = alloc((size_t)4096 * H);
    float* xs1  = alloc((size_t)2048 * H);
    float* xs2  = alloc((size_t)1024 * H);
    float* xa   = alloc((size_t)N0 * H);
    float* xb   = alloc((size_t)N0 * H);
    float* cat  = alloc((size_t)N0 * 2 * H);
    float* A1p  = alloc((size_t)2048 * 2048);
    float* A2p  = alloc((size_t)1024 * 1024);
    float* A3p  = alloc((size_t)512 * 512);
    int* perm0 = (int*)(ws + off); off += 2048;
    int* perm1 = (int*)(ws + off); off += 1024;
    int* perm2 = (int*)(ws + off); off += 512;

    // ---- Down path -------------------------------------------------------
    // x = relu(gcn(A, x, Wd0)), saved as xs0
    launch_gcn(stream, A_in, 4096, x_in, 128, H, Wd0, bd0, xs0, true, dinv, z);

    // level 0: augment(A_in) -> pool by score(xs0,p1) -> gcn Wd1 -> xs1
    {
        int n = 4096, k = 2048;
        launch_augment(stream, A_in, n, Aaug);
        score_kernel<<<(n + 255) / 256, 256, 0, stream>>>(xs0, p1, n, H, score);
        topk_kernel<<<1, 1024, 0, stream>>>(score, n, k, perm0, sval);
        gather_rows_kernel<<<(k * H + 255) / 256, 256, 0, stream>>>(
            xs0, perm0, sval, H, k, xa);
        gather_adj_kernel<<<(int)(((size_t)k * k + 255) / 256), 256, 0, stream>>>(
            Aaug, n, perm0, k, A1p);
        launch_gcn(stream, A1p, k, xa, H, H, Wd1, bd1, xs1, true, dinv, z);
    }
    // level 1
    {
        int n = 2048, k = 1024;
        launch_augment(stream, A1p, n, Aaug);
        score_kernel<<<(n + 255) / 256, 256, 0, stream>>>(xs1, p2, n, H, score);
        topk_kernel<<<1, 1024, 0, stream>>>(score, n, k, perm1, sval);
        gather_rows_kernel<<<(k * H + 255) / 256, 256, 0, stream>>>(
            xs1, perm1, sval, H, k, xa);
        gather_adj_kernel<<<(int)(((size_t)k * k + 255) / 256), 256, 0, stream>>>(
            Aaug, n, perm1, k, A2p);
        launch_gcn(stream, A2p, k, xa, H, H, Wd2, bd2, xs2, true, dinv, z);
    }
    // level 2 (deepest): result in xb (512 x 256)
    {
        int n = 1024, k = 512;
        launch_augment(stream, A2p, n, Aaug);
        score_kernel<<<(n + 255) / 256, 256, 0, stream>>>(xs2, p3, n, H, score);
        topk_kernel<<<1, 1024, 0, stream>>>(score, n, k, perm2, sval);
        gather_rows_kernel<<<(k * H + 255) / 256, 256, 0, stream>>>(
            xs2, perm2, sval, H, k, xa);
        gather_adj_kernel<<<(int)(((size_t)k * k + 255) / 256), 256, 0, stream>>>(
            Aaug, n, perm2, k, A3p);
        launch_gcn(stream, A3p, k, xa, H, H, Wd3, bd3, xb, true, dinv, z);
    }

    // ---- Up path ---------------------------------------------------------
    // i=0: res=xs2 (1024), perm2 (512), x=xb -> xa
    {
        int n = 1024, k = 512;
        concat_init_kernel<<<(int)(((size_t)n * 2 * H + 255) / 256), 256, 0,
                             stream>>>(xs2, n, H, cat);
        scatter_rows_kernel<<<(k * H + 255) / 256, 256, 0, stream>>>(xb, perm2,
                                                                     k, H, cat);
        launch_gcn(stream, A2p, n, cat, 2 * H, H, Wu0, bu0, xa, true, dinv, z);
    }
    // i=1: res=xs1 (2048), perm1 (1024), x=xa -> xb
    {
        int n = 2048, k = 1024;
        concat_init_kernel<<<(int)(((size_t)n * 2 * H + 255) / 256), 256, 0,
                             stream>>>(xs1, n, H, cat);
        scatter_rows_kernel<<<(k * H + 255) / 256, 256, 0, stream>>>(xa, perm1,
                                                                     k, H, cat);
        launch_gcn(stream, A1p, n, cat, 2 * H, H, Wu1, bu1, xb, true, dinv, z);
    }
    // i=2: res=xs0 (4096), perm0 (2048), x=xb -> d_out (no relu)
    {
        int n = 4096, k = 2048;
        concat_init_kernel<<<(int)(((size_t)n * 2 * H + 255) / 256), 256, 0,
                             stream>>>(xs0, n, H, cat);
        scatter_rows_kernel<<<(k * H + 255) / 256, 256, 0, stream>>>(xb, perm0,
                                                                     k, H, cat);
        launch_gcn(stream, A_in, n, cat, 2 * H, 128, Wu2, bu2, out_x, false,
                   dinv, z);
    }

    // Second output: the original adjacency (overwrites the Aaug scratch).
    hipMemcpyAsync(Aaug, A_in, (size_t)N0 * N0 * sizeof(float),
                   hipMemcpyDeviceToDevice, stream);
}